// MPC_86260123174396
// MI455X (gfx1250) — compile-verified
//
#include <hip/hip_runtime.h>
#include <stdint.h>

typedef __attribute__((ext_vector_type(2))) float v2f;
typedef __attribute__((ext_vector_type(8))) float v8f;

#define T_STEPS 64
#define B_SZ    2048
#define NS      16
#define NC      8
#define NSC     24
#define CSTR    28   // C_s row stride (floats); 28*4=112B, multiple of 16B for async B128

__device__ __forceinline__ v8f wmma4(v2f a, v2f b, v8f c) {
  // V_WMMA_F32_16X16X4_F32 : D(16x16,f32) = A(16x4,f32) * B(4x16,f32) + C
  return __builtin_amdgcn_wmma_f32_16x16x4_f32(
      /*neg_a=*/false, a, /*neg_b=*/false, b,
      /*c_mod=*/(short)0, c, /*reuse_a=*/false, /*reuse_b=*/false);
}

__device__ __forceinline__ void wait_async0() {
  asm volatile("s_wait_asynccnt 0x0" ::: "memory");
}

// Async-stage a 24x24 f32 matrix (row-contiguous, 96B rows) into LDS with row
// stride CSTR floats. 144 16B chunks = 4 full wave-ops + one 16-lane op.
__device__ __forceinline__ void stage_C_async(const float* __restrict__ g,
                                              float* l, int lane) {
  uint32_t lb = (uint32_t)(uintptr_t)l;
  #pragma unroll
  for (int i = 0; i < 4; ++i) {
    int c = i * 32 + lane;
    uint32_t laddr = lb + (uint32_t)((c / 6) * (CSTR * 4) + (c % 6) * 16);
    int gaddr = c * 16;
    asm volatile("global_load_async_to_lds_b128 %0, %1, %2"
                 :: "v"(laddr), "v"(gaddr), "s"(g) : "memory");
  }
  if (lane < 16) {
    int c = 128 + lane;
    uint32_t laddr = lb + (uint32_t)((c / 6) * (CSTR * 4) + (c % 6) * 16);
    int gaddr = c * 16;
    asm volatile("global_load_async_to_lds_b128 %0, %1, %2"
                 :: "v"(laddr), "v"(gaddr), "s"(g) : "memory");
  }
}

// Async-stage a 16x24 f32 matrix into LDS with row stride 32 floats (128B).
// 96 16B chunks = 3 full wave-ops.
__device__ __forceinline__ void stage_F_async(const float* __restrict__ g,
                                              float* l, int lane) {
  uint32_t lb = (uint32_t)(uintptr_t)l;
  #pragma unroll
  for (int i = 0; i < 3; ++i) {
    int c = i * 32 + lane;
    uint32_t laddr = lb + (uint32_t)((c / 6) * 128 + (c % 6) * 16);
    int gaddr = c * 16;
    asm volatile("global_load_async_to_lds_b128 %0, %1, %2"
                 :: "v"(laddr), "v"(gaddr), "s"(g) : "memory");
  }
}

__global__ __launch_bounds__(32)
void lqr_step_kernel(const float* __restrict__ x_init,
                     const float* __restrict__ Cg_all,
                     const float* __restrict__ cg_all,
                     const float* __restrict__ Fg_all,
                     const float* __restrict__ fg_all,
                     const float* __restrict__ cx_all,
                     const float* __restrict__ cu_all,
                     float* __restrict__ out) {
  // One wave32 == one batch element.
  __shared__ float Kall[T_STEPS][NC * 17];   // K (8x16) + k (8) per t
  __shared__ float C_s[2][NSC * CSTR];       // double-buffered 24x24
  __shared__ float F_s[2][NS * 32];          // double-buffered 16x24, pad cols 24..31 = 0
  __shared__ float V_s[16 * 16];
  __shared__ float W_s[16 * 32];             // V*F, padded
  __shared__ float Q_s[32 * 33];             // Qt padded 32x32, stride 33
  __shared__ float K_s[16 * 16];             // K padded (rows 8..15 = 0)
  __shared__ float Z_s[16 * 16];             // Qux + Quu*K (rows >=8 garbage, unused)
  __shared__ float q_s[32];
  __shared__ float v_s[16];
  __shared__ float xu_s[32];
  __shared__ float kt_s[8];
  __shared__ float t1_s[8];
  __shared__ float x_s[16];

  const int lane = threadIdx.x;  // 0..31
  const int b    = blockIdx.x;   // batch element
  const int ml   = lane & 15;
  const int hi   = lane >> 4;

  // zero-init carried state and pad regions
  for (int i = lane; i < 16 * 16; i += 32) { V_s[i] = 0.f; K_s[i] = 0.f; }
  for (int i = lane; i < 2 * 16 * 32; i += 32) F_s[0][i] = 0.f;
  if (lane < 16) v_s[lane] = 0.f;
  __syncthreads();  // drain zeroing stores before async engine writes

  // prologue: prefetch t = 63 into buffer (63 & 1) = 1
  {
    const size_t tb0 = (size_t)(T_STEPS - 1) * B_SZ + b;
    stage_C_async(Cg_all + tb0 * (NSC * NSC), C_s[1], lane);
    stage_F_async(Fg_all + tb0 * (NS * NSC),  F_s[1], lane);
  }

  // ---------------- backward Riccati scan ----------------
  for (int t = T_STEPS - 1; t >= 0; --t) {
    const size_t tb = (size_t)t * B_SZ + b;
    const int buf = t & 1;
    const float* Cb = C_s[buf];
    const float* Fb = F_s[buf];

    wait_async0();       // staged C/F for this step are in LDS
    __syncthreads();

    // prefetch next (t-1) step into the other buffer, overlapped with compute
    if (t > 0) {
      const size_t tbn = tb - B_SZ;
      stage_C_async(Cg_all + tbn * (NSC * NSC), C_s[buf ^ 1], lane);
      stage_F_async(Fg_all + tbn * (NS * NSC),  F_s[buf ^ 1], lane);
    }

    if (lane < NSC)
      xu_s[lane] = (lane < NS) ? cx_all[tb * NS + lane] : cu_all[tb * NC + (lane - NS)];
    __syncthreads();

    // q = c + C*xu + F^T v   (24-vector, lane-parallel)
    if (lane < NSC) {
      float acc = cg_all[tb * NSC + lane];
      #pragma unroll
      for (int j = 0; j < NSC; ++j) acc += Cb[lane * CSTR + j] * xu_s[j];
      #pragma unroll
      for (int k = 0; k < NS; ++k) acc += Fb[k * 32 + lane] * v_s[k];
      q_s[lane] = acc;
    }

    // W = V * F  (16 x 24, two 16x16 tiles, K=16 -> 4 wmma each)
    #pragma unroll
    for (int tj = 0; tj < 2; ++tj) {
      v8f acc = {};
      #pragma unroll
      for (int kb = 0; kb < 16; kb += 4) {
        v2f a, bb;
        a.x  = V_s[ml * 16 + kb + 2 * hi + 0];
        a.y  = V_s[ml * 16 + kb + 2 * hi + 1];
        bb.x = Fb[(kb + 2 * hi + 0) * 32 + tj * 16 + ml];
        bb.y = Fb[(kb + 2 * hi + 1) * 32 + tj * 16 + ml];
        acc = wmma4(a, bb, acc);
      }
      #pragma unroll
      for (int r = 0; r < 8; ++r) W_s[(r + 8 * hi) * 32 + tj * 16 + ml] = acc[r];
    }
    __syncthreads();

    // Q = C + F^T * W  (4 tiles; seed loads branchless: clamped addr + cndmask)
    #pragma unroll
    for (int ti = 0; ti < 2; ++ti) {
      #pragma unroll
      for (int tj = 0; tj < 2; ++tj) {
        v8f acc;
        #pragma unroll
        for (int r = 0; r < 8; ++r) {
          int M = ti * 16 + r + 8 * hi, N = tj * 16 + ml;
          int Mc = (M < NSC) ? M : (NSC - 1);
          int Nc = (N < NSC) ? N : (NSC - 1);
          float cv = Cb[Mc * CSTR + Nc];          // always in-bounds ds_load
          acc[r] = (M < NSC && N < NSC) ? cv : 0.f;  // v_cndmask, no branch
        }
        #pragma unroll
        for (int kb = 0; kb < 16; kb += 4) {
          v2f a, bb;  // A = F^T : element(m,k) = F[k][m]
          a.x  = Fb[(kb + 2 * hi + 0) * 32 + ti * 16 + ml];
          a.y  = Fb[(kb + 2 * hi + 1) * 32 + ti * 16 + ml];
          bb.x = W_s[(kb + 2 * hi + 0) * 32 + tj * 16 + ml];
          bb.y = W_s[(kb + 2 * hi + 1) * 32 + tj * 16 + ml];
          acc = wmma4(a, bb, acc);
        }
        #pragma unroll
        for (int r = 0; r < 8; ++r)
          Q_s[(ti * 16 + r + 8 * hi) * 33 + tj * 16 + ml] = acc[r];
      }
    }
    __syncthreads();

    // Solve Quu X = [Qux | qu] via wave-parallel Gauss-Jordan (SPD, no pivot).
    // Lane j owns column j of the 8x25 augmented system; branchless gather.
    const float* srcp = (lane < 8)  ? &Q_s[16 * 33 + 16 + lane]
                      : (lane < 24) ? &Q_s[16 * 33 + (lane - 8)]
                                    : &q_s[16];
    const int sstr = (lane < 24) ? 33 : 1;
    float a[8];
    #pragma unroll
    for (int i = 0; i < 8; ++i) a[i] = srcp[i * sstr];
    #pragma unroll
    for (int k = 0; k < 8; ++k) {
      float ck[8];
      #pragma unroll
      for (int i = 0; i < 8; ++i) ck[i] = __shfl(a[i], k, 32);
      float pinv = 1.0f / ck[k];
      a[k] *= pinv;
      #pragma unroll
      for (int i = 0; i < 8; ++i)
        if (i != k) a[i] -= ck[i] * a[k];
    }
    // K = -Quu^{-1} Qux, k = -Quu^{-1} qu  (lane 24's column IS the k slot, col 16)
    if (lane >= 8 && lane <= 24) {
      int col = lane - 8;  // 0..16
      #pragma unroll
      for (int i = 0; i < 8; ++i) Kall[t][i * 17 + col] = -a[i];
    }
    if (lane >= 8 && lane < 24) {
      #pragma unroll
      for (int i = 0; i < 8; ++i) K_s[i * 16 + (lane - 8)] = -a[i];
    } else if (lane == 24) {
      #pragma unroll
      for (int i = 0; i < 8; ++i) kt_s[i] = -a[i];
    }
    __syncthreads();

    // Z = Qux + Quu*K  (rows >= 8 garbage, killed by K^T zero rows later)
    {
      v8f acc;
      #pragma unroll
      for (int r = 0; r < 8; ++r) acc[r] = Q_s[(16 + r + 8 * hi) * 33 + ml];
      #pragma unroll
      for (int kb = 0; kb < 16; kb += 4) {
        v2f a2, bb;  // A = Quu_pad
        a2.x = Q_s[(16 + ml) * 33 + 16 + kb + 2 * hi + 0];
        a2.y = Q_s[(16 + ml) * 33 + 16 + kb + 2 * hi + 1];
        bb.x = K_s[(kb + 2 * hi + 0) * 16 + ml];
        bb.y = K_s[(kb + 2 * hi + 1) * 16 + ml];
        acc = wmma4(a2, bb, acc);
      }
      #pragma unroll
      for (int r = 0; r < 8; ++r) Z_s[(r + 8 * hi) * 16 + ml] = acc[r];
    }
    __syncthreads();

    // Vnew = Qxx + Qxu*K + K^T*Z
    {
      v8f acc;
      #pragma unroll
      for (int r = 0; r < 8; ++r) acc[r] = Q_s[(r + 8 * hi) * 33 + ml];  // Qxx
      #pragma unroll
      for (int kb = 0; kb < 16; kb += 4) {  // + Qxu * K
        v2f a2, bb;
        a2.x = Q_s[ml * 33 + 16 + kb + 2 * hi + 0];
        a2.y = Q_s[ml * 33 + 16 + kb + 2 * hi + 1];
        bb.x = K_s[(kb + 2 * hi + 0) * 16 + ml];
        bb.y = K_s[(kb + 2 * hi + 1) * 16 + ml];
        acc = wmma4(a2, bb, acc);
      }
      #pragma unroll
      for (int kb = 0; kb < 16; kb += 4) {  // + K^T * Z
        v2f a2, bb;  // A = K^T : element(m,k) = K[k][m]
        a2.x = K_s[(kb + 2 * hi + 0) * 16 + ml];
        a2.y = K_s[(kb + 2 * hi + 1) * 16 + ml];
        bb.x = Z_s[(kb + 2 * hi + 0) * 16 + ml];
        bb.y = Z_s[(kb + 2 * hi + 1) * 16 + ml];
        acc = wmma4(a2, bb, acc);
      }
      __syncthreads();
      #pragma unroll
      for (int r = 0; r < 8; ++r) V_s[(r + 8 * hi) * 16 + ml] = acc[r];
    }

    // vnew = qx + Qxu*kt + K^T*(qu + Quu*kt)
    if (lane < 8) {
      float acc = q_s[16 + lane];
      #pragma unroll
      for (int j = 0; j < 8; ++j) acc += Q_s[(16 + lane) * 33 + 16 + j] * kt_s[j];
      t1_s[lane] = acc;
    }
    __syncthreads();
    if (lane < 16) {
      float acc = q_s[lane];
      #pragma unroll
      for (int j = 0; j < 8; ++j) acc += Q_s[lane * 33 + 16 + j] * kt_s[j];
      #pragma unroll
      for (int i = 0; i < 8; ++i) acc += K_s[i * 16 + lane] * t1_s[i];
      v_s[lane] = acc;
    }
    __syncthreads();
  }

  // ---------------- forward rollout ----------------
  // Note: backward t=0 left F(t=0) staged in F_s[0]; parity matches t&1 below.
  if (lane < NS) x_s[lane] = x_init[(size_t)b * NS + lane];
  float* out_x = out;
  float* out_u = out + (size_t)T_STEPS * B_SZ * NS;

  for (int t = 0; t < T_STEPS; ++t) {
    const size_t tb = (size_t)t * B_SZ + b;
    const float* Fb = F_s[t & 1];

    wait_async0();
    __syncthreads();
    if (t + 1 < T_STEPS) {  // prefetch F(t+1) into the other buffer
      stage_F_async(Fg_all + (tb + B_SZ) * (NS * NSC), F_s[(t + 1) & 1], lane);
    }

    if (lane < NS) {
      float xv = x_s[lane];
      out_x[tb * NS + lane] = xv;               // new_x[t] = x before update
      xu_s[lane] = xv;
      q_s[lane] = xv - cx_all[tb * NS + lane];  // dx (reuse q_s)
    }
    __syncthreads();
    if (lane < NC) {
      float u = cu_all[tb * NC + lane] + Kall[t][lane * 17 + 16];
      #pragma unroll
      for (int j = 0; j < NS; ++j) u += Kall[t][lane * 17 + j] * q_s[j];
      xu_s[NS + lane] = u;
      out_u[tb * NC + lane] = u;
    }
    __syncthreads();
    float xn = 0.f;
    if (lane < NS) {
      xn = fg_all[tb * NS + lane];
      #pragma unroll
      for (int j = 0; j < NSC; ++j) xn += Fb[lane * 32 + j] * xu_s[j];
    }
    __syncthreads();
    if (lane < NS) x_s[lane] = xn;
  }
}

extern "C" void kernel_launch(void* const* d_in, const int* in_sizes, int n_in,
                              void* d_out, int out_size, void* d_ws, size_t ws_size,
                              hipStream_t stream) {
  const float* x_init = (const float*)d_in[0];
  const float* C      = (const float*)d_in[1];
  const float* c      = (const float*)d_in[2];
  const float* F      = (const float*)d_in[3];
  const float* f      = (const float*)d_in[4];
  const float* cx     = (const float*)d_in[5];
  const float* cu     = (const float*)d_in[6];
  lqr_step_kernel<<<B_SZ, 32, 0, stream>>>(x_init, C, c, F, f, cx, cu, (float*)d_out);
}